// Message_9560597201508
// MI455X (gfx1250) — compile-verified
//
#include <hip/hip_runtime.h>
#include <math.h>

typedef __attribute__((ext_vector_type(2))) float v2f;
typedef __attribute__((ext_vector_type(8))) float v8f;

#define SDIM 128          // memory / hidden dim
#define TDIM 128          // time-encode dim
#define DEDIM 128         // edge feature dim
#define MDIM 512          // GRU input dim = 2*S + T + DE
#define G3 384            // 3*S gate dim
#define XPAD 516          // 512 + 4 pad (keeps 8B align, kills LDS bank conflicts)
#define HPAD 132
#define GPAD 388

__global__ void init_last_kernel(int* last, int n) {
    int i = blockIdx.x * blockDim.x + threadIdx.x;
    if (i < n) last[i] = -1;
}

__global__ void last_edge_kernel(const int* __restrict__ src, const int* __restrict__ dst,
                                 int* last_r, int* last_f, int E) {
    int eid = blockIdx.x * blockDim.x + threadIdx.x;
    if (eid < E) {
        atomicMax(&last_f[dst[eid]], eid);   // forward graph: reduce onto items
        atomicMax(&last_r[src[eid]], eid);   // reverse graph: reduce onto users
    }
}

__device__ __forceinline__ v8f wmma4(v2f a, v2f b, v8f c) {
    // V_WMMA_F32_16X16X4_F32 : D[16x16] = A[16x4] * B[4x16] + C  (fp32, wave32)
    return __builtin_amdgcn_wmma_f32_16x16x4_f32(false, a, false, b, (short)0, c, false, false);
}

// One block = 16 nodes. 256 threads = 8 waves; wave w owns output tiles w*3..w*3+2
// (16 features each, 24 tiles = 384 gate outputs).
__global__ __launch_bounds__(256) void tgn_gru_kernel(
    const float* __restrict__ si, const float* __restrict__ sj,
    const float* __restrict__ t,  const float* __restrict__ e,
    const int* __restrict__ src,  const int* __restrict__ dst,
    const float* __restrict__ Wi, const float* __restrict__ Wh,
    const float* __restrict__ bi, const float* __restrict__ bh,
    const float* __restrict__ bf,
    const int* __restrict__ last,     // last_f (items) or last_r (users)
    const float* __restrict__ hmem,   // sj (items) or si (users)
    float* __restrict__ out,
    int nnodes, int is_item)
{
    __shared__ __align__(16) float Xs[16][XPAD];   // GRU input x  [16 x 512]
    __shared__ __align__(16) float Hs[16][HPAD];   // hidden h     [16 x 128]
    __shared__ __align__(16) float Gs[16][GPAD];   // [0,256): gi+gh sums (r,z); [256,384): gi only (n)
    __shared__ __align__(16) float GHn[16][HPAD];  // gh n-part    [16 x 128]
    __shared__ int   s_eid[16], s_i1[16], s_i2[16];
    __shared__ float s_t[16];

    const int tid = threadIdx.x;
    const int nb  = blockIdx.x * 16;

    if (tid < 16) {
        int node = nb + tid;
        int eid  = (node < nnodes) ? last[node] : -1;
        s_eid[tid] = eid;
        int se = (eid >= 0) ? src[eid] : 0;
        int de = (eid >= 0) ? dst[eid] : 0;
        // items (fwd mail):  x = [si[src], sj[dst], te, e]
        // users (rev mail):  x = [sj[dst], si[src], te, e]
        s_i1[tid] = is_item ? se : de;
        s_i2[tid] = is_item ? de : se;
        s_t[tid]  = (eid >= 0) ? t[eid] : 0.0f;
    }
    __syncthreads();

    // ---- load H tile (coalesced) ----
    for (int i = tid; i < 16 * SDIM; i += 256) {
        int m = i >> 7, f = i & (SDIM - 1);
        int node = nb + m;
        Hs[m][f] = (node < nnodes) ? hmem[(size_t)node * SDIM + f] : 0.0f;
    }
    // ---- build X tile: gathers + time encode ----
    const float* m1 = is_item ? si : sj;
    const float* m2 = is_item ? sj : si;
    for (int m = 0; m < 16; m++) {
        int eid = s_eid[m];
        for (int f = tid; f < MDIM; f += 256) {
            float v;
            if (eid < 0)              v = 0.0f;
            else if (f < SDIM)        v = m1[(size_t)s_i1[m] * SDIM + f];
            else if (f < 2 * SDIM)    v = m2[(size_t)s_i2[m] * SDIM + (f - SDIM)];
            else if (f < 2 * SDIM + TDIM) v = cosf(s_t[m] * bf[f - 2 * SDIM]);
            else                      v = e[(size_t)eid * DEDIM + (f - 2 * SDIM - TDIM)];
            Xs[m][f] = v;
        }
    }
    __syncthreads();

    // ---- GEMM phase: gi = x @ Wi^T + bi ; gh = h @ Wh^T + bh  via fp32 WMMA ----
    const int lane = tid & 31;
    const int wave = tid >> 5;
    const int nn   = lane & 15;   // A: row M ; B: col N
    const int hi   = lane >> 4;   // K group selector (K = 2*hi + reg)
    for (int j = 0; j < 3; j++) {
        const int obase = (wave * 3 + j) * 16;
        const int o = obase + nn;

        v8f acc = {0.f, 0.f, 0.f, 0.f, 0.f, 0.f, 0.f, 0.f};
        const float* wrow = Wi + (size_t)o * MDIM + 2 * hi;
        #pragma unroll 8
        for (int kb = 0; kb < MDIM; kb += 4) {
            v2f a = *(const v2f*)&Xs[nn][kb + 2 * hi];
            v2f b = *(const v2f*)&wrow[kb];
            acc = wmma4(a, b, acc);
        }

        v8f acch = {0.f, 0.f, 0.f, 0.f, 0.f, 0.f, 0.f, 0.f};
        const float* hrow = Wh + (size_t)o * SDIM + 2 * hi;
        #pragma unroll 8
        for (int kb = 0; kb < SDIM; kb += 4) {
            v2f a = *(const v2f*)&Hs[nn][kb + 2 * hi];
            v2f b = *(const v2f*)&hrow[kb];
            acch = wmma4(a, b, acch);
        }

        const float bsum = bi[o] + bh[o];
        if (obase < 2 * SDIM) {
            // r / z gates only ever need gi+gh
            #pragma unroll
            for (int r = 0; r < 8; r++)
                Gs[r + 8 * hi][o] = acc[r] + acch[r] + bsum;
        } else {
            // n gate needs gi and gh separately (gin + r*ghn)
            const float bii = bi[o], bhh = bh[o];
            #pragma unroll
            for (int r = 0; r < 8; r++) {
                Gs[r + 8 * hi][o]              = acc[r]  + bii;
                GHn[r + 8 * hi][o - 2 * SDIM]  = acch[r] + bhh;
            }
        }
    }
    __syncthreads();

    // ---- elementwise GRU combine ----
    for (int i = tid; i < 16 * SDIM; i += 256) {
        int m = i >> 7, f = i & (SDIM - 1);
        int node = nb + m;
        if (node >= nnodes) continue;
        float rs  = Gs[m][f];                 // gir + ghr (+biases)
        float zs  = Gs[m][SDIM + f];          // giz + ghz (+biases)
        float gin = Gs[m][2 * SDIM + f];      // gin + bi
        float ghn = GHn[m][f];                // ghn + bh
        float r  = 1.0f / (1.0f + expf(-rs));
        float z  = 1.0f / (1.0f + expf(-zs));
        float nv = tanhf(gin + r * ghn);
        float h  = Hs[m][f];
        out[(size_t)node * SDIM + f] = (1.0f - z) * nv + z * h;
    }
}

extern "C" void kernel_launch(void* const* d_in, const int* in_sizes, int n_in,
                              void* d_out, int out_size, void* d_ws, size_t ws_size,
                              hipStream_t stream) {
    (void)n_in; (void)out_size; (void)ws_size;
    const float* si = (const float*)d_in[0];
    const float* sj = (const float*)d_in[1];
    const float* t  = (const float*)d_in[2];
    const float* e  = (const float*)d_in[3];
    const int*   src = (const int*)d_in[4];
    const int*   dst = (const int*)d_in[5];
    const float* Wi = (const float*)d_in[6];
    const float* Wh = (const float*)d_in[7];
    const float* bi = (const float*)d_in[8];
    const float* bh = (const float*)d_in[9];
    const float* bf = (const float*)d_in[10];

    const int n_user = in_sizes[0] / SDIM;
    const int n_item = in_sizes[1] / SDIM;
    const int E      = in_sizes[2];

    int* last_r = (int*)d_ws;          // [n_user]
    int* last_f = last_r + n_user;     // [n_item]

    init_last_kernel<<<(n_user + n_item + 255) / 256, 256, 0, stream>>>(last_r, n_user + n_item);
    last_edge_kernel<<<(E + 255) / 256, 256, 0, stream>>>(src, dst, last_r, last_f, E);

    float* out_u = (float*)d_out;                         // si_new
    float* out_i = out_u + (size_t)n_user * SDIM;         // sj_new

    // users: reverse graph, h = si
    tgn_gru_kernel<<<(n_user + 15) / 16, 256, 0, stream>>>(
        si, sj, t, e, src, dst, Wi, Wh, bi, bh, bf, last_r, si, out_u, n_user, 0);
    // items: forward graph, h = sj
    tgn_gru_kernel<<<(n_item + 15) / 16, 256, 0, stream>>>(
        si, sj, t, e, src, dst, Wi, Wh, bi, bh, bf, last_f, sj, out_i, n_item, 1);
}